// HMM_70205535420588
// MI455X (gfx1250) — compile-verified
//
#include <hip/hip_runtime.h>
#include <hip/hip_bf16.h>

#define BB 8
#define SS 512
#define HH 1024
#define VV 32000
#define VSPLIT 4
#define VCHUNK (VV / VSPLIT)   // 8000

typedef __attribute__((ext_vector_type(2))) float v2f;
typedef __attribute__((ext_vector_type(8))) float v8f;

// ---------------- workspace layout (32-bit words) ----------------
// cnt16      : [0,       512000)  u32  counts[b][v], 16 rows, rows 8..15 = 0 pad
// sumexp_em  : [512000,  513024)  f32  sum_v exp(em[h,v])
// dot        : [513024,  521216)  f32  dot[b][h] = sum_v cnt[b][v]*em[h,v]
// sumexp_c   : [521216,  522240)  f32  sum_j exp(tm[j,i]-lse_row[j])
// lse_row_tm : [522240,  523264)  f32
#define WS_CNT      0
#define WS_SUMEXP   512000
#define WS_DOT      513024
#define WS_SUMEXPC  521216
#define WS_LSEROW   522240
#define WS_ZERO_N   522240   // cnt16 + sumexp_em + dot + sumexp_c start at zero

__global__ void hmm_zero(unsigned* __restrict__ ws, int nwords) {
    int i = blockIdx.x * blockDim.x + threadIdx.x;
    if (i < nwords) ws[i] = 0u;
}

__global__ void hmm_count(const int* __restrict__ ids, unsigned* __restrict__ cnt) {
    int t = blockIdx.x * blockDim.x + threadIdx.x;   // 0..B*S-1
    int b = t / SS;
    int id = ids[t];
    atomicAdd(&cnt[(size_t)b * VV + id], 1u);
}

// lse_row[j] = log(sum_i exp(tm[j,i]))  (tm ~ N(0,1): unstabilized exp is safe)
__global__ void hmm_tm_rows(const float* __restrict__ tm, float* __restrict__ lse_row) {
    __shared__ float red[256];
    const int j = blockIdx.x, t = threadIdx.x;
    const float4 x = *(const float4*)(tm + (size_t)j * HH + t * 4);
    red[t] = __expf(x.x) + __expf(x.y) + __expf(x.z) + __expf(x.w);
    __syncthreads();
    for (int s = 128; s > 0; s >>= 1) {
        if (t < s) red[t] += red[t + s];
        __syncthreads();
    }
    if (t == 0) lse_row[j] = __logf(red[0]);
}

// sumexp_c[i] += sum_j exp(tm[j,i] - lse_row[j])  over this block's j slice
__global__ void hmm_tm_cols(const float* __restrict__ tm,
                            const float* __restrict__ lse_row,
                            float* __restrict__ sumexp_c) {
    const int i  = blockIdx.x * blockDim.x + threadIdx.x;   // column, coalesced
    const int j0 = blockIdx.y * (HH / 8);
    float s = 0.f;
    for (int j = j0; j < j0 + HH / 8; ++j)
        s += __expf(tm[(size_t)j * HH + i] - lse_row[j]);
    atomicAdd(&sumexp_c[i], s);
}

// Stream em once: per 16-row tile, accumulate
//   dot[b][h] += sum_v cnt[b][v] * em[h][v]      (WMMA f32 16x16x4, rows 8..15 zero-pad)
//   sumexp[h] += sum_v exp(em[h][v])
__global__ void __launch_bounds__(256)
hmm_em(const float* __restrict__ em, const unsigned* __restrict__ cnt,
       float* __restrict__ dot, float* __restrict__ sumexp_em) {
    __shared__ float s_red[8 * 256];
    __shared__ float s_se[16];

    const int tid  = threadIdx.x;
    const int wave = tid >> 5;
    const int lane = tid & 31;
    const int n    = lane & 15;          // N index (h within tile) == A-matrix row (b)
    const int kb   = (lane >> 4) << 1;   // K base: lanes 0-15 -> K{0,1}, 16-31 -> K{2,3}
    const int h0   = blockIdx.x * 16;
    const int vbase = blockIdx.y * VCHUNK;

    if (tid < 16) s_se[tid] = 0.f;
    __syncthreads();

    const float*    brow = em  + (size_t)(h0 + n) * VV + kb;  // B: em[h0+n][v+kb..+1]
    const unsigned* arow = cnt + (size_t)n * VV + kb;         // A: cnt16[n][v+kb..+1]

    v8f acc0 = {}, acc1 = {};
    float se = 0.f;
    const int vend = vbase + VCHUNK;
    // 2x unrolled, two independent accumulators -> breaks the D->C WMMA chain
    for (int v = vbase + (wave << 2); v < vend; v += 8 * 8) {
        const float2 bv0 = *(const float2*)(brow + v);
        const float2 bv1 = *(const float2*)(brow + v + 32);
        const uint2  cu0 = *(const uint2*)(arow + v);
        const uint2  cu1 = *(const uint2*)(arow + v + 32);
        v2f av0 = {(float)cu0.x, (float)cu0.y};
        v2f av1 = {(float)cu1.x, (float)cu1.y};
        v2f bb0 = {bv0.x, bv0.y};
        v2f bb1 = {bv1.x, bv1.y};
        acc0 = __builtin_amdgcn_wmma_f32_16x16x4_f32(false, av0, false, bb0,
                                                     (short)0, acc0, false, false);
        acc1 = __builtin_amdgcn_wmma_f32_16x16x4_f32(false, av1, false, bb1,
                                                     (short)0, acc1, false, false);
        se += __expf(bv0.x) + __expf(bv0.y) + __expf(bv1.x) + __expf(bv1.y);
    }

    atomicAdd(&s_se[n], se);             // ds_add_f32 per-row exp-sum
#pragma unroll
    for (int r = 0; r < 8; ++r)
        s_red[wave * 256 + r * 32 + lane] = acc0[r] + acc1[r];
    __syncthreads();

    // combine the 8 waves' accumulators; D layout: VGPR r, lanes<16 -> M=r, N=lane
    float s = 0.f;
#pragma unroll
    for (int w = 0; w < 8; ++w) s += s_red[w * 256 + tid];
    const int r = tid >> 5, l = tid & 31;
    if (l < 16)   // M = r (upper lanes would be M=r+8 -> zero padding rows, dropped)
        atomicAdd(&dot[(size_t)r * HH + h0 + l], s);
    if (tid < 16)
        atomicAdd(&sumexp_em[h0 + tid], s_se[tid]);
}

// out = -mean_b LSE_h( p_ls[h] + (S-1)*c[h] + dot[b][h] - S*lse_em[h] )
__global__ void __launch_bounds__(1024)
hmm_final(const float* __restrict__ p, const float* __restrict__ sumexp_c,
          const float* __restrict__ sumexp_em, const float* __restrict__ dot,
          float* __restrict__ out) {
    __shared__ float red[1024];
    const int i = threadIdx.x;
    const float pv = p[i];

    // stable LSE of p
    red[i] = pv; __syncthreads();
    for (int s = 512; s > 0; s >>= 1) { if (i < s) red[i] = fmaxf(red[i], red[i + s]); __syncthreads(); }
    const float mp = red[0]; __syncthreads();
    red[i] = __expf(pv - mp); __syncthreads();
    for (int s = 512; s > 0; s >>= 1) { if (i < s) red[i] += red[i + s]; __syncthreads(); }
    const float lse_p = mp + __logf(red[0]); __syncthreads();

    const float base = pv - lse_p + (float)(SS - 1) * __logf(sumexp_c[i])
                     - (float)SS * __logf(sumexp_em[i]);

    float ll_sum = 0.f;
    for (int b = 0; b < BB; ++b) {
        const float a = base + dot[(size_t)b * HH + i];
        red[i] = a; __syncthreads();
        for (int s = 512; s > 0; s >>= 1) { if (i < s) red[i] = fmaxf(red[i], red[i + s]); __syncthreads(); }
        const float mv = red[0]; __syncthreads();
        red[i] = __expf(a - mv); __syncthreads();
        for (int s = 512; s > 0; s >>= 1) { if (i < s) red[i] += red[i + s]; __syncthreads(); }
        if (i == 0) ll_sum += mv + __logf(red[0]);
        __syncthreads();
    }
    if (i == 0) out[0] = -ll_sum / (float)BB;
}

extern "C" void kernel_launch(void* const* d_in, const int* in_sizes, int n_in,
                              void* d_out, int out_size, void* d_ws, size_t ws_size,
                              hipStream_t stream) {
    const int*   ids = (const int*)d_in[0];
    // d_in[1] = do_em (ignored, == 0 in the pure forward)
    const float* em  = (const float*)d_in[2];
    const float* tm  = (const float*)d_in[3];
    const float* p   = (const float*)d_in[4];
    float* out = (float*)d_out;

    unsigned* ws_u      = (unsigned*)d_ws;
    unsigned* cnt       = ws_u + WS_CNT;
    float*    sumexp_em = (float*)d_ws + WS_SUMEXP;
    float*    dot       = (float*)d_ws + WS_DOT;
    float*    sumexp_c  = (float*)d_ws + WS_SUMEXPC;
    float*    lse_row   = (float*)d_ws + WS_LSEROW;

    hmm_zero<<<(WS_ZERO_N + 255) / 256, 256, 0, stream>>>(ws_u, WS_ZERO_N);
    hmm_count<<<(BB * SS) / 256, 256, 0, stream>>>(ids, cnt);
    hmm_tm_rows<<<HH, 256, 0, stream>>>(tm, lse_row);
    hmm_tm_cols<<<dim3(HH / 256, 8), 256, 0, stream>>>(tm, lse_row, sumexp_c);
    hmm_em<<<dim3(HH / 16, VSPLIT), 256, 0, stream>>>(em, cnt, dot, sumexp_em);
    hmm_final<<<1, 1024, 0, stream>>>(p, sumexp_c, sumexp_em, dot, out);
}